// lossFunc_11046655885808
// MI455X (gfx1250) — compile-verified
//
#include <hip/hip_runtime.h>

// Problem constants (B=64, T=200, Q=4000 from the reference setup)
#define NB 64
#define NT 200
#define NQ 4000
#define NS 199              // S = T-1
#define NST 13              // ceil(199/16) s-tiles
#define SPAD 208            // NST*16
#define NWAVE 8             // waves per block in fused kernel
#define QCHUNK (NQ / NWAVE) // 500, multiple of 4

// Workspace layout (floats)
#define ARRSZ (NB * SPAD)         // 13312
#define OFF_PC 0
#define OFF_PT (ARRSZ)
#define OFF_PE (2 * ARRSZ)
#define OFF_DS (3 * ARRSZ)
#define OFF_AD (4 * ARRSZ)
#define OFF_KD (5 * ARRSZ)        // NST*NB + NB = 896 partials
#define NKD (NST * NB + NB)
#define OFF_LB (OFF_KD + NKD)     // 64 per-batch loss partials

typedef __attribute__((ext_vector_type(2))) float v2f;
typedef __attribute__((ext_vector_type(8))) float v8f;

// sigmoid with hardware reciprocal (v_rcp_f32, ~1 ulp)
__device__ __forceinline__ float sigm(float x) {
  return __builtin_amdgcn_rcpf(1.0f + __expf(-x));
}

__device__ __forceinline__ float wave_sum(float v) {
#pragma unroll
  for (int off = 16; off > 0; off >>= 1) v += __shfl_xor(v, off, 32);
  return v;
}

// ---------------------------------------------------------------------------
// Fused streaming kernel: one pass over logits (t=0..198) + batch (t=1..199).
// Computes: KD partial sums, pc/pt/pe einsum via V_WMMA_F32_16X16X4_F32,
// dsum (has_q) and adiff (for `a`). Branchless inner loop: padded lanes
// (s >= 199, only in tile 12) are clamped to row 0; they pollute only
// diagonal entries / ws slots that the finalize never reads, and their KD
// contribution is zeroed after the loop.
//
// Per element we compute u = e^{-x} once, then:
//   sigma(x)  = rcp(1 + u)          (A-fragment value)
//   sigma(2x) = rcp(1 + u^2)        (KD value; e^{-2x} = u^2, exact identity)
// Both rcps depend only on u -> parallel trans issue, no extra multiplies.
//
// Grid: (NST, NB), 256 threads = 8 waves; wave w owns q in [w*500, w*500+500).
// ---------------------------------------------------------------------------
__global__ __launch_bounds__(256) void k_fused(const float* __restrict__ lc,
                                               const float* __restrict__ lt,
                                               const float* __restrict__ le,
                                               const float* __restrict__ bt,
                                               float* __restrict__ ws) {
  const int stile = blockIdx.x;
  const int b = blockIdx.y;
  const int tid = threadIdx.x;
  const int wave = tid >> 5;
  const int lane = tid & 31;
  const int s0 = stile * 16;
  const int row = s0 + (lane & 15);        // s index this lane feeds (M / N)
  const int qoff = (lane >> 4) << 1;       // lanes 0-15 -> K{0,1}; 16-31 -> K{2,3}
  const bool vr = row < NS;
  const int rowc = vr ? row : 0;           // clamp padded lanes to a valid row

  // uniform per-block base pointers (land in SGPRs) + 32-bit per-lane offsets
  const float* __restrict__ lcb = lc + (size_t)b * (NT * NQ);
  const float* __restrict__ ltb = lt + (size_t)b * (NT * NQ);
  const float* __restrict__ leb = le + (size_t)b * (NT * NQ);
  const float* __restrict__ btb = bt + (size_t)b * (NT * 2 * NQ);
  const int qbeg = wave * QCHUNK;
  int li = rowc * NQ + qbeg + qoff;              // logits at time s
  int bi = (rowc + 1) * (2 * NQ) + qbeg + qoff;  // batch at time s+1

  v8f Cc = {0.f, 0.f, 0.f, 0.f, 0.f, 0.f, 0.f, 0.f};
  v8f Ct = Cc, Ce = Cc;
  float kdl = 0.f, dsl = 0.f, adl = 0.f;

  for (int it = 0; it < QCHUNK / 4; ++it) {
    const v2f xc = *(const v2f*)(lcb + li);
    const v2f xt = *(const v2f*)(ltb + li);
    const v2f xe = *(const v2f*)(leb + li);
    const v2f b0 = *(const v2f*)(btb + bi);
    const v2f b1 = *(const v2f*)(btb + bi + NQ);
    li += 4;
    bi += 4;

    // u = exp(-x) for all six logit values
    const float uc0 = __expf(-xc.x), uc1 = __expf(-xc.y);
    const float ut0 = __expf(-xt.x), ut1 = __expf(-xt.y);
    const float ue0 = __expf(-xe.x), ue1 = __expf(-xe.y);

    v2f ac, at_, ae, bd;
    ac.x = __builtin_amdgcn_rcpf(1.0f + uc0);
    ac.y = __builtin_amdgcn_rcpf(1.0f + uc1);
    at_.x = __builtin_amdgcn_rcpf(1.0f + ut0);
    at_.y = __builtin_amdgcn_rcpf(1.0f + ut1);
    ae.x = __builtin_amdgcn_rcpf(1.0f + ue0);
    ae.y = __builtin_amdgcn_rcpf(1.0f + ue1);

    // sigma(2x) = 1/(1+u^2)
    const float c20 = __builtin_amdgcn_rcpf(fmaf(uc0, uc0, 1.0f));
    const float c21 = __builtin_amdgcn_rcpf(fmaf(uc1, uc1, 1.0f));
    const float t20 = __builtin_amdgcn_rcpf(fmaf(ut0, ut0, 1.0f));
    const float t21 = __builtin_amdgcn_rcpf(fmaf(ut1, ut1, 1.0f));
    const float e20 = __builtin_amdgcn_rcpf(fmaf(ue0, ue0, 1.0f));
    const float e21 = __builtin_amdgcn_rcpf(fmaf(ue1, ue1, 1.0f));

    kdl += fabsf(e20 - c20) + fabsf(e21 - c21) +
           fabsf(e20 - t20) + fabsf(e21 - t21);

    bd.x = b0.x + b1.x; bd.y = b0.y + b1.y;      // d1 (valid baked into batch)
    dsl += bd.x + bd.y;
    adl += (b0.x - b1.x) + (b0.y - b1.y);

    // D = A x B + C, exact f32; diagonal is the per-row dot product
    Cc = __builtin_amdgcn_wmma_f32_16x16x4_f32(false, ac, false, bd, (short)0, Cc, false, false);
    Ct = __builtin_amdgcn_wmma_f32_16x16x4_f32(false, at_, false, bd, (short)0, Ct, false, false);
    Ce = __builtin_amdgcn_wmma_f32_16x16x4_f32(false, ae, false, bd, (short)0, Ce, false, false);
  }

  kdl = vr ? kdl : 0.f;  // padded lanes must not contribute to the KD sum

  // combine the two K-half lanes for dsum/adiff: row m = lane m + lane m+16
  dsl += __shfl_xor(dsl, 16, 32);
  adl += __shfl_xor(adl, 16, 32);

  // diagonal extraction: (M,M) is at lane M (vgpr M) for M<8,
  // and at lane M+16 (vgpr M-8) for M>=8.
  const bool have = (lane < 8) || (lane >= 24);
  const int rsel = (lane < 8) ? lane : (lane - 24);
  float dc = 0.f, dt = 0.f, de = 0.f;
#pragma unroll
  for (int r = 0; r < 8; ++r) {
    if (have && rsel == r) { dc = Cc[r]; dt = Ct[r]; de = Ce[r]; }
  }

  __shared__ float spc[NWAVE][16], spt[NWAVE][16], spe[NWAVE][16];
  __shared__ float sds[NWAVE][16], sad[NWAVE][16], skd[NWAVE];
  const int mrow = (lane < 8) ? lane : (lane - 16);
  if (have) { spc[wave][mrow] = dc; spt[wave][mrow] = dt; spe[wave][mrow] = de; }
  if (lane < 16) { sds[wave][lane] = dsl; sad[wave][lane] = adl; }
  kdl = wave_sum(kdl);
  if (lane == 0) skd[wave] = kdl;
  __syncthreads();

  if (tid < 16) {
    float a0 = 0.f, a1 = 0.f, a2 = 0.f, a3 = 0.f, a4 = 0.f;
#pragma unroll
    for (int w = 0; w < NWAVE; ++w) {
      a0 += spc[w][tid]; a1 += spt[w][tid]; a2 += spe[w][tid];
      a3 += sds[w][tid]; a4 += sad[w][tid];
    }
    const int o = b * SPAD + s0 + tid;
    ws[OFF_PC + o] = a0; ws[OFF_PT + o] = a1; ws[OFF_PE + o] = a2;
    ws[OFF_DS + o] = a3; ws[OFF_AD + o] = a4;
  }
  if (tid == 0) {
    float k = 0.f;
#pragma unroll
    for (int w = 0; w < NWAVE; ++w) k += skd[w];
    ws[OFF_KD + b * NST + stile] = k;
  }
}

// ---------------------------------------------------------------------------
// KD for the last time row (t = T-1), not touched by the fused pass.
// ---------------------------------------------------------------------------
__global__ __launch_bounds__(256) void k_kdlast(const float* __restrict__ lc,
                                                const float* __restrict__ lt,
                                                const float* __restrict__ le,
                                                float* __restrict__ ws) {
  const int b = blockIdx.x, tid = threadIdx.x;
  const size_t base = ((size_t)b * NT + (NT - 1)) * NQ;
  float kdl = 0.f;
  for (int q = tid; q < NQ; q += 256) {
    const float uc = __expf(-lc[base + q]);
    const float ut = __expf(-lt[base + q]);
    const float ue = __expf(-le[base + q]);
    const float c2 = __builtin_amdgcn_rcpf(fmaf(uc, uc, 1.0f));
    const float t2 = __builtin_amdgcn_rcpf(fmaf(ut, ut, 1.0f));
    const float e2 = __builtin_amdgcn_rcpf(fmaf(ue, ue, 1.0f));
    kdl += fabsf(e2 - c2) + fabsf(e2 - t2);
  }
  __shared__ float sm[8];
  kdl = wave_sum(kdl);
  if ((tid & 31) == 0) sm[tid >> 5] = kdl;
  __syncthreads();
  if (tid == 0) {
    float t = 0.f;
#pragma unroll
    for (int w = 0; w < 8; ++w) t += sm[w];
    ws[OFF_KD + NST * NB + b] = t;
  }
}

__device__ __forceinline__ float bce1(float p, float a) {
  p = fminf(fmaxf(p, 1e-7f), 1.f - 1e-7f);
  return -(a * logf(p) + (1.f - a) * log1pf(-p));
}

// ---------------------------------------------------------------------------
// Per-batch finalize: suffix cummax mask == (s <= last index with has_q),
// BCE of the three dot results, prediction / ground_truth, per-b loss partial.
// ---------------------------------------------------------------------------
__global__ __launch_bounds__(256) void k_final(float* __restrict__ ws,
                                               float* __restrict__ out) {
  const int b = blockIdx.x, tid = threadIdx.x;
  const int lane = tid & 31, wave = tid >> 5;
  __shared__ int smx[8];
  __shared__ int Ls;
  __shared__ float smf[8];

  int Ll = -1;
  for (int s = tid; s < NS; s += 256)
    if (ws[OFF_DS + b * SPAD + s] > 0.f) Ll = s;
#pragma unroll
  for (int off = 16; off > 0; off >>= 1) {
    const int o = __shfl_xor(Ll, off, 32);
    Ll = (o > Ll) ? o : Ll;
  }
  if (lane == 0) smx[wave] = Ll;
  __syncthreads();
  if (tid == 0) {
    int m = -1;
#pragma unroll
    for (int w = 0; w < 8; ++w) m = (smx[w] > m) ? smx[w] : m;
    Ls = m;
  }
  __syncthreads();
  const int L = Ls;
  const float cnt = fmaxf((float)(L + 1), 1.0f);

  float* pred = out + 2;
  float* gt = out + 2 + NB * NS;
  float acc = 0.f;
  for (int s = tid; s < NS; s += 256) {
    const int o = b * SPAD + s;
    const float mask = (s <= L) ? 1.f : 0.f;
    const float a = (ws[OFF_AD + o] > 0.5f) ? 1.f : 0.f;
    const float pcv = ws[OFF_PC + o], ptv = ws[OFF_PT + o], pev = ws[OFF_PE + o];
    if (mask > 0.f) acc += bce1(ptv, a) + bce1(pcv, a) + bce1(pev, a);
    pred[b * NS + s] = (ptv + pcv + pev) * (1.0f / 3.0f) * mask;
    gt[b * NS + s] = a * mask;
  }
  acc = wave_sum(acc);
  if (lane == 0) smf[wave] = acc;
  __syncthreads();
  if (tid == 0) {
    float t = 0.f;
#pragma unroll
    for (int w = 0; w < 8; ++w) t += smf[w];
    ws[OFF_LB + b] = t / cnt;
  }
}

// ---------------------------------------------------------------------------
// Final scalars: loss = sum of per-batch partials; loss_kd = 0.1 * sum(kd).
// ---------------------------------------------------------------------------
__global__ __launch_bounds__(256) void k_scalars(const float* __restrict__ ws,
                                                 float* __restrict__ out) {
  const int tid = threadIdx.x, lane = tid & 31, wave = tid >> 5;
  float s1 = 0.f, s2 = 0.f;
  for (int i = tid; i < NB; i += 256) s1 += ws[OFF_LB + i];
  for (int i = tid; i < NKD; i += 256) s2 += ws[OFF_KD + i];
  __shared__ float m1[8], m2[8];
  s1 = wave_sum(s1);
  s2 = wave_sum(s2);
  if (lane == 0) { m1[wave] = s1; m2[wave] = s2; }
  __syncthreads();
  if (tid == 0) {
    float t1 = 0.f, t2 = 0.f;
#pragma unroll
    for (int w = 0; w < 8; ++w) { t1 += m1[w]; t2 += m2[w]; }
    out[0] = t1;
    out[1] = 0.1f * t2;
  }
}

extern "C" void kernel_launch(void* const* d_in, const int* in_sizes, int n_in,
                              void* d_out, int out_size, void* d_ws, size_t ws_size,
                              hipStream_t stream) {
  (void)in_sizes; (void)n_in; (void)out_size; (void)ws_size;
  const float* lc = (const float*)d_in[0];
  const float* lt = (const float*)d_in[1];
  const float* le = (const float*)d_in[2];
  const float* bt = (const float*)d_in[3];
  float* ws = (float*)d_ws;
  float* out = (float*)d_out;

  k_fused<<<dim3(NST, NB), 256, 0, stream>>>(lc, lt, le, bt, ws);
  k_kdlast<<<dim3(NB), 256, 0, stream>>>(lc, lt, le, ws);
  k_final<<<dim3(NB), 256, 0, stream>>>(ws, out);
  k_scalars<<<1, 256, 0, stream>>>(ws, out);
}